// MinamoTopoModel_87857851007597
// MI455X (gfx1250) — compile-verified
//
#include <hip/hip_runtime.h>
#include <hip/hip_bf16.h>

typedef float v2f __attribute__((ext_vector_type(2)));
typedef float v8f __attribute__((ext_vector_type(8)));

#define NN 100000
#define EE 3200000
#define GG 64
// EMB=16, HID=32, OUT=16, MLP=8

// ---------------- zero ----------------
__global__ void k_zero(float* p, long long n) {
    long long i = (long long)blockIdx.x * blockDim.x + threadIdx.x;
    long long stride = (long long)gridDim.x * blockDim.x;
    for (; i < n; i += stride) p[i] = 0.0f;
}

// ---------------- degree (in-edges; self-loop folded into rsqrt) ----------------
__global__ void k_degree(const int* __restrict__ ei, float* __restrict__ deg) {
    long long e = (long long)blockIdx.x * blockDim.x + threadIdx.x;
    if (e >= EE) return;
    int d = ei[EE + e];
    atomicAdd(&deg[d], 1.0f);
}

__global__ void k_dis(float* __restrict__ deg) {
    int i = blockIdx.x * blockDim.x + threadIdx.x;
    if (i >= NN) return;
    deg[i] = rsqrtf(deg[i] + 1.0f);   // +1 for the self-loop; always > 0
}

// ---------------- layer1: t = emb[x_nodes] @ W1   (16 -> 32), WMMA f32 16x16x4 ----------------
// one wave handles a tile of 16 nodes; N=100000 = 6250 full tiles -> EXEC all-1s
__global__ void k_xw1(const int* __restrict__ xn, const float* __restrict__ emb,
                      const float* __restrict__ W1, float* __restrict__ t, int nTiles) {
    int wave = (int)(((long long)blockIdx.x * blockDim.x + threadIdx.x) >> 5);
    int lane = threadIdx.x & 31;
    if (wave >= nTiles) return;                 // uniform per wave
    int base = wave * 16;
    int m    = lane & 15;                       // M index (A rows) and N index (B/D cols)
    int half = lane >> 4;                       // upper lanes carry K+2/K+3 (A,B) and M+8 (D)
    int row  = xn[base + m];                    // embedding row 0..31
    const float* arow = emb + row * 16;

    v8f acc0 = {};   // columns 0..15 of W1
    v8f acc1 = {};   // columns 16..31 of W1
#pragma unroll
    for (int i = 0; i < 4; ++i) {
        int k0 = 4 * i + 2 * half;
        v2f a;  a.x  = arow[k0];                 a.y  = arow[k0 + 1];
        v2f b0; b0.x = W1[k0 * 32 + m];          b0.y = W1[(k0 + 1) * 32 + m];
        v2f b1; b1.x = W1[k0 * 32 + 16 + m];     b1.y = W1[(k0 + 1) * 32 + 16 + m];
        acc0 = __builtin_amdgcn_wmma_f32_16x16x4_f32(false, a, false, b0, (short)0, acc0, false, false);
        acc1 = __builtin_amdgcn_wmma_f32_16x16x4_f32(false, a, false, b1, (short)0, acc1, false, false);
    }
#pragma unroll
    for (int r = 0; r < 8; ++r) {
        int M = r + 8 * half;
        t[(long long)(base + M) * 32 + m]      = acc0[r];
        t[(long long)(base + M) * 32 + 16 + m] = acc1[r];
    }
}

// ---------------- edge scatter, 32 components: one wave per edge ----------------
__global__ void k_scatter32(const int* __restrict__ ei, const float* __restrict__ dis,
                            const float* __restrict__ x, float* __restrict__ acc) {
    long long tid = (long long)blockIdx.x * blockDim.x + threadIdx.x;
    long long e = tid >> 5;
    int c = (int)(tid & 31);
    if (e >= EE) return;
    int s = ei[e];
    int d = ei[EE + e];
    float nrm = dis[s] * dis[d];
    atomicAdd(&acc[(long long)d * 32 + c], x[(long long)s * 32 + c] * nrm);
}

// ---------------- self-loop + bias + relu:  t <- relu(acc + t*dis^2 + b1) ----------------
__global__ void k_fin1(const float* __restrict__ acc, const float* __restrict__ dis,
                       const float* __restrict__ b1, float* __restrict__ t) {
    long long tid = (long long)blockIdx.x * blockDim.x + threadIdx.x;
    if (tid >= (long long)NN * 32) return;
    int i = (int)(tid >> 5);
    int c = (int)(tid & 31);
    float ds = dis[i];
    float v = acc[tid] + t[tid] * ds * ds + b1[c];
    t[tid] = fmaxf(v, 0.0f);
}

// ---------------- layer2: u = t @ W2   (32 -> 16), WMMA f32 16x16x4, K=32 ----------------
__global__ void k_xw2(const float* __restrict__ t, const float* __restrict__ W2,
                      float* __restrict__ u, int nTiles) {
    int wave = (int)(((long long)blockIdx.x * blockDim.x + threadIdx.x) >> 5);
    int lane = threadIdx.x & 31;
    if (wave >= nTiles) return;
    int base = wave * 16;
    int m    = lane & 15;
    int half = lane >> 4;
    const float* arow = t + (long long)(base + m) * 32;

    v8f acc = {};
#pragma unroll
    for (int i = 0; i < 8; ++i) {
        int k0 = 4 * i + 2 * half;
        v2f a; a.x = arow[k0];            a.y = arow[k0 + 1];
        v2f b; b.x = W2[k0 * 16 + m];     b.y = W2[(k0 + 1) * 16 + m];
        acc = __builtin_amdgcn_wmma_f32_16x16x4_f32(false, a, false, b, (short)0, acc, false, false);
    }
#pragma unroll
    for (int r = 0; r < 8; ++r) {
        int M = r + 8 * half;
        u[(long long)(base + M) * 16 + m] = acc[r];
    }
}

// ---------------- edge scatter, 16 components: two edges per wave ----------------
__global__ void k_scatter16(const int* __restrict__ ei, const float* __restrict__ dis,
                            const float* __restrict__ x, float* __restrict__ acc) {
    long long tid = (long long)blockIdx.x * blockDim.x + threadIdx.x;
    long long e = tid >> 4;
    int c = (int)(tid & 15);
    if (e >= EE) return;
    int s = ei[e];
    int d = ei[EE + e];
    float nrm = dis[s] * dis[d];
    atomicAdd(&acc[(long long)d * 16 + c], x[(long long)s * 16 + c] * nrm);
}

// ---------------- self-loop + bias + batched pooling (sums + counts) ----------------
__global__ void k_fin2_pool(const float* __restrict__ acc, const float* __restrict__ dis,
                            const float* __restrict__ b2, const float* __restrict__ u,
                            const int* __restrict__ batch,
                            float* __restrict__ pooled, float* __restrict__ cnt) {
    long long tid = (long long)blockIdx.x * blockDim.x + threadIdx.x;
    if (tid >= (long long)NN * 16) return;
    int i = (int)(tid >> 4);
    int c = (int)(tid & 15);
    float ds = dis[i];
    float v = acc[tid] + u[tid] * ds * ds + b2[c];
    int g = batch[i];
    atomicAdd(&pooled[g * 16 + c], v);
    if (c == 0) atomicAdd(&cnt[g], 1.0f);
}

// ---------------- final: out = (pooled/cnt) @ fcW + fcb  [64 x 8] ----------------
__global__ void k_fc(const float* __restrict__ pooled, const float* __restrict__ cnt,
                     const float* __restrict__ fcW, const float* __restrict__ fcb,
                     float* __restrict__ out) {
    int tid = blockIdx.x * blockDim.x + threadIdx.x;
    if (tid >= GG * 8) return;
    int g = tid >> 3;
    int mcol = tid & 7;
    float inv = 1.0f / fmaxf(cnt[g], 1.0f);
    float s = 0.0f;
#pragma unroll
    for (int c = 0; c < 16; ++c) s += pooled[g * 16 + c] * inv * fcW[c * 8 + mcol];
    out[tid] = s + fcb[mcol];
}

extern "C" void kernel_launch(void* const* d_in, const int* in_sizes, int n_in,
                              void* d_out, int out_size, void* d_ws, size_t ws_size,
                              hipStream_t stream) {
    const int*   xn    = (const int*)d_in[0];    // [N]
    const int*   ei    = (const int*)d_in[1];    // [2,E]
    const int*   batch = (const int*)d_in[2];    // [N]
    const float* emb   = (const float*)d_in[3];  // [32,16]
    const float* W1    = (const float*)d_in[4];  // [16,32]
    const float* b1    = (const float*)d_in[5];  // [32]
    const float* W2    = (const float*)d_in[6];  // [32,16]
    const float* b2    = (const float*)d_in[7];  // [16]
    const float* fcW   = (const float*)d_in[8];  // [16,8]
    const float* fcb   = (const float*)d_in[9];  // [8]
    float* out = (float*)d_out;                  // [64,8]

    // workspace layout (floats). zero-initialized region comes first.
    float* deg    = (float*)d_ws;                 // N   (becomes dis in place)
    float* acc1   = deg + NN;                     // N*32
    float* acc2   = acc1 + (long long)NN * 32;    // N*16
    float* pooled = acc2 + (long long)NN * 16;    // G*16
    float* cnt    = pooled + GG * 16;             // G
    float* t      = cnt + GG;                     // N*32 (XW1 -> relu'd h1)
    float* u      = t + (long long)NN * 32;       // N*16 (h1 @ W2)

    const long long zeroCount = (long long)NN * (1 + 32 + 16) + GG * 17;
    k_zero<<<4096, 256, 0, stream>>>(deg, zeroCount);

    k_degree<<<(EE + 255) / 256, 256, 0, stream>>>(ei, deg);
    k_dis<<<(NN + 255) / 256, 256, 0, stream>>>(deg);

    const int nTiles = NN / 16;                   // 6250 exact
    const int wmmaBlocks = (nTiles * 32 + 255) / 256;
    k_xw1<<<wmmaBlocks, 256, 0, stream>>>(xn, emb, W1, t, nTiles);

    k_scatter32<<<(int)(((long long)EE * 32 + 255) / 256), 256, 0, stream>>>(ei, deg, t, acc1);
    k_fin1<<<(int)(((long long)NN * 32 + 255) / 256), 256, 0, stream>>>(acc1, deg, b1, t);

    k_xw2<<<wmmaBlocks, 256, 0, stream>>>(t, W2, u, nTiles);

    k_scatter16<<<(int)(((long long)EE * 16 + 255) / 256), 256, 0, stream>>>(ei, deg, u, acc2);
    k_fin2_pool<<<(int)(((long long)NN * 16 + 255) / 256), 256, 0, stream>>>(acc2, deg, b2, u, batch, pooled, cnt);

    k_fc<<<2, 256, 0, stream>>>(pooled, cnt, fcW, fcb, out);

    (void)in_sizes; (void)n_in; (void)out_size; (void)ws_size;
}